// BatchHardTripletLoss_33775622815971
// MI455X (gfx1250) — compile-verified
//
#include <hip/hip_runtime.h>
#include <hip/hip_bf16.h>
#include <math.h>

typedef __attribute__((ext_vector_type(2))) float v2f;
typedef __attribute__((ext_vector_type(4))) float v4f;
typedef __attribute__((ext_vector_type(8))) float v8f;

#define N_ROWS 8192
#define DIM    128
#define MARGIN 0.3f
#define BIGV   1e30f

#define LDS_STRIDE 132          // 128 floats + 4-dword pad -> conflict-free frags

// ---------------- Kernel 1: row squared norms ----------------
__global__ __launch_bounds__(256) void sqnorm_kernel(const float* __restrict__ x,
                                                     float* __restrict__ sq) {
    int row = blockIdx.x * blockDim.x + threadIdx.x;
    if (row >= N_ROWS) return;
    const v4f* p = (const v4f*)(x + (size_t)row * DIM);
    float acc = 0.f;
#pragma unroll
    for (int i = 0; i < DIM / 4; ++i) {
        v4f v = p[i];
        acc += v.x * v.x + v.y * v.y + v.z * v.z + v.w * v.w;
    }
    sq[row] = acc;
}

// ---------------- Kernel 2: fused WMMA Gram + masked hardest pos/neg ----------------
// 8 waves/block, one 16-row strip per wave; all waves share the J-tile stream.
// J tiles are cooperatively staged into double-buffered LDS with coalesced
// global_load_b128, then each wave reads its B fragments via conflict-free
// ds_load_b64 (bank = 4*idx + const). A fragments live in 64 VGPRs.
// V_WMMA_F32_16X16X4_F32 fragment layout:
//   lanes 0-15:  row = idx, float2 at k0+0 ; lanes 16-31: row = idx, float2 at k0+2
// C/D layout: VGPR v -> row M = v + 8*half, lane%16 -> column N.
// Hardest-pos/neg tracked in squared-distance space; clamp+sqrt deferred.
__global__ __launch_bounds__(256) void triplet_strip_kernel(
    const float* __restrict__ x, const int* __restrict__ lab,
    const float* __restrict__ sq,
    float* __restrict__ row_loss, float* __restrict__ row_valid) {

    __shared__ float sbuf[2][16 * LDS_STRIDE];   // 2 x 8448 B

    const int tid  = threadIdx.x;
    const int lane = tid & 31;
    const int wave = tid >> 5;
    const int strip = blockIdx.x * 8 + wave;     // 512 strips
    const int I = strip * 16;
    const int half = lane >> 4;
    const int idx  = lane & 15;

    // Staging assignment: thread t fills float4 slots t and t+256 of the tile.
    const int r0 = tid >> 5,         c0 = (tid & 31) * 4;         // rows 0..7
    const int r1 = (tid + 256) >> 5, c1 = c0;                     // rows 8..15

    // Preload the A fragments for this strip: 32 K-steps of 4.
    v2f afrag[32];
    const float* arow = x + (size_t)(I + idx) * DIM + 2 * half;
#pragma unroll
    for (int ks = 0; ks < 32; ++ks)
        afrag[ks] = *(const v2f*)(arow + 4 * ks);

    float rsq[8]; int rlab[8];
#pragma unroll
    for (int v = 0; v < 8; ++v) {
        int m = v + 8 * half;
        rsq[v]  = sq[I + m];
        rlab[v] = lab[I + m];
    }

    float accp[8], accn[8];   // running hardest pos (max d^2) / neg (min d^2)
#pragma unroll
    for (int v = 0; v < 8; ++v) { accp[v] = -BIGV; accn[v] = BIGV; }

    // Prologue: stage tile J=0 into buffer 0 (coalesced b128 loads).
    {
        v4f t0 = *(const v4f*)(x + (size_t)r0 * DIM + c0);
        v4f t1 = *(const v4f*)(x + (size_t)r1 * DIM + c1);
        *(v4f*)&sbuf[0][r0 * LDS_STRIDE + c0] = t0;
        *(v4f*)&sbuf[0][r1 * LDS_STRIDE + c1] = t1;
    }
    __syncthreads();

    for (int J = 0; J < N_ROWS; J += 16) {
        const int cur = (J >> 4) & 1;
        const int nxt = cur ^ 1;
        const int Jn  = J + 16;

        // Issue next tile's global loads early (latency hidden by compute).
        v4f t0, t1;
        if (Jn < N_ROWS) {
            t0 = *(const v4f*)(x + (size_t)(Jn + r0) * DIM + c0);
            t1 = *(const v4f*)(x + (size_t)(Jn + r1) * DIM + c1);
        }

        // Compute 16x16 Gram tile from LDS (conflict-free ds_load_b64).
        const float* bbase = &sbuf[cur][idx * LDS_STRIDE + 2 * half];
        v8f c = {};
#pragma unroll
        for (int ks = 0; ks < 32; ++ks) {
            v2f b = *(const v2f*)(bbase + 4 * ks);
            c = __builtin_amdgcn_wmma_f32_16x16x4_f32(
                    /*neg_a=*/false, afrag[ks], /*neg_b=*/false, b,
                    /*c_mod=*/(short)0, c, /*reuse_a=*/false, /*reuse_b=*/false);
        }

        float csq  = sq[J + idx];
        int   clab = lab[J + idx];
        int   gj   = J + idx;            // global column of this lane
#pragma unroll
        for (int v = 0; v < 8; ++v) {
            int gi = I + v + 8 * half;   // global row of this element
            float d2 = fmaf(-2.0f, c[v], rsq[v] + csq);   // may be slightly <0
            bool same = (rlab[v] == clab);
            float pcand = (same && (gi != gj)) ? d2 : -BIGV;
            float ncand = (!same) ? d2 : BIGV;
            accp[v] = fmaxf(accp[v], pcand);
            accn[v] = fminf(accn[v], ncand);
        }

        // Publish next tile. Write target == buffer read in the previous iteration;
        // the barrier below (and the previous one) makes this safe.
        if (Jn < N_ROWS) {
            *(v4f*)&sbuf[nxt][r0 * LDS_STRIDE + c0] = t0;
            *(v4f*)&sbuf[nxt][r1 * LDS_STRIDE + c1] = t1;
        }
        __syncthreads();
    }

    // Reduce across the 16 lanes of each half (xor masks < 16 stay in-half).
#pragma unroll
    for (int v = 0; v < 8; ++v) {
#pragma unroll
        for (int off = 8; off >= 1; off >>= 1) {
            accp[v] = fmaxf(accp[v], __shfl_xor(accp[v], off, 32));
            accn[v] = fminf(accn[v], __shfl_xor(accn[v], off, 32));
        }
    }
    if (idx == 0) {   // lanes 0 and 16 hold rows I+v and I+8+v
#pragma unroll
        for (int v = 0; v < 8; ++v) {
            int row = I + v + 8 * half;
            float hp2 = accp[v], hn2 = accn[v];
            bool valid = (hp2 > -1e29f) && (hn2 < 1e29f);
            float hp = sqrtf(fmaxf(hp2, 0.0f));   // deferred clamp + sqrt
            float hn = sqrtf(fmaxf(hn2, 0.0f));
            float pr = fmaxf(hp - hn + MARGIN, 0.0f);
            row_loss[row]  = valid ? pr : 0.0f;
            row_valid[row] = valid ? 1.0f : 0.0f;
        }
    }
}

// ---------------- Kernel 3: deterministic final reduction ----------------
__global__ __launch_bounds__(256) void reduce_kernel(const float* __restrict__ rl,
                                                     const float* __restrict__ rv,
                                                     float* __restrict__ out) {
    __shared__ float ss[256];
    __shared__ float sc[256];
    float s = 0.f, c = 0.f;
    for (int i = threadIdx.x; i < N_ROWS; i += 256) { s += rl[i]; c += rv[i]; }
    ss[threadIdx.x] = s; sc[threadIdx.x] = c;
    __syncthreads();
    for (int off = 128; off > 0; off >>= 1) {
        if (threadIdx.x < off) {
            ss[threadIdx.x] += ss[threadIdx.x + off];
            sc[threadIdx.x] += sc[threadIdx.x + off];
        }
        __syncthreads();
    }
    if (threadIdx.x == 0)
        out[0] = (sc[0] > 0.f) ? ss[0] / fmaxf(sc[0], 1.f) : 0.f;
}

extern "C" void kernel_launch(void* const* d_in, const int* in_sizes, int n_in,
                              void* d_out, int out_size, void* d_ws, size_t ws_size,
                              hipStream_t stream) {
    const float* x   = (const float*)d_in[0];   // [8192, 128] fp32
    const int*   lab = (const int*)d_in[1];     // [8192] int
    float* out = (float*)d_out;

    float* sq        = (float*)d_ws;            // 8192 floats
    float* row_loss  = sq + N_ROWS;             // 8192 floats
    float* row_valid = row_loss + N_ROWS;       // 8192 floats

    sqnorm_kernel<<<N_ROWS / 256, 256, 0, stream>>>(x, sq);

    // 512 strips of 16 rows; 8 waves (256 threads) per block -> 64 blocks.
    triplet_strip_kernel<<<(N_ROWS / 16) / 8, 256, 0, stream>>>(x, lab, sq,
                                                                row_loss, row_valid);

    reduce_kernel<<<1, 256, 0, stream>>>(row_loss, row_valid, out);
}